// EnhancedCGMNMemory_42202348650970
// MI455X (gfx1250) — compile-verified
//
#include <hip/hip_runtime.h>
#include <math.h>

typedef __attribute__((ext_vector_type(2))) float v2f;
typedef __attribute__((ext_vector_type(8))) float v8f;

#define TOKENS   8192
#define IN_DIM   1024
#define FDIM     48
#define HODE     128
#define MSLOTS   8192
#define HDIM     256
#define KBIG     48
#define KBASE    32
#define DTSTEP   0.5f
#define LB_TH    0.7f

__device__ __forceinline__ float gelu_exact(float x) {
    return 0.5f * x * (1.0f + erff(x * 0.70710678118654752440f));
}

__device__ __forceinline__ v8f wmma_f32(v2f a, v2f b, v8f c) {
    return __builtin_amdgcn_wmma_f32_16x16x4_f32(false, a, false, b,
                                                 (short)0, c, false, false);
}

// ---------------------------------------------------------------------------
// K1: man = x @ w_proj + b_proj  (fp32 WMMA). One wave owns a 16-row M-tile
// and keeps THREE accumulators (n0 = 0,16,32) so each A fragment is reused
// for 3 WMMAs -> x is streamed from HBM exactly once (32 MB, ~1.4us@23.3TB/s).
// ---------------------------------------------------------------------------
__global__ void __launch_bounds__(128) proj_gemm(const float* __restrict__ x,
                                                 const float* __restrict__ wp,
                                                 const float* __restrict__ bp,
                                                 float* __restrict__ man) {
    const int lane = threadIdx.x & 31;
    const int wave = threadIdx.x >> 5;
    const int half = lane >> 4;
    const int lm   = lane & 15;
    const int m0 = (blockIdx.x * 4 + wave) * 16;
    v8f acc0 = {0.f,0.f,0.f,0.f,0.f,0.f,0.f,0.f};
    v8f acc1 = acc0, acc2 = acc0;
    for (int k = 0; k < IN_DIM; k += 4) {
        const int ka = k + 2 * half;
        v2f a = *(const v2f*)(x + (size_t)(m0 + lm) * IN_DIM + ka);
        const float* wr0 = wp + (size_t)ka * FDIM + lm;        // row ka
        const float* wr1 = wr0 + FDIM;                         // row ka+1
        v2f b0, b1, b2;
        b0.x = wr0[0];  b0.y = wr1[0];
        b1.x = wr0[16]; b1.y = wr1[16];
        b2.x = wr0[32]; b2.y = wr1[32];
        acc0 = wmma_f32(a, b0, acc0);
        acc1 = wmma_f32(a, b1, acc1);
        acc2 = wmma_f32(a, b2, acc2);
    }
#pragma unroll
    for (int r = 0; r < 8; ++r) {
        const int row = m0 + r + 8 * half;
        float* mr = man + (size_t)row * FDIM;
        mr[lm]      = acc0[r] + bp[lm];
        mr[lm + 16] = acc1[r] + bp[lm + 16];
        mr[lm + 32] = acc2[r] + bp[lm + 32];
    }
}

// ---------------------------------------------------------------------------
// K2: per-row LayerNorm(48) + exact GELU, in place
// ---------------------------------------------------------------------------
__global__ void __launch_bounds__(256) ln_gelu_f(float* __restrict__ man,
                                                 const float* __restrict__ g,
                                                 const float* __restrict__ b) {
    const int row = blockIdx.x * blockDim.x + threadIdx.x;
    float v[FDIM];
    float mu = 0.f;
#pragma unroll
    for (int f = 0; f < FDIM; ++f) { v[f] = man[(size_t)row * FDIM + f]; mu += v[f]; }
    mu *= (1.0f / FDIM);
    float var = 0.f;
#pragma unroll
    for (int f = 0; f < FDIM; ++f) { float d = v[f] - mu; var += d * d; }
    var *= (1.0f / FDIM);
    const float rs = rsqrtf(var + 1e-5f);
#pragma unroll
    for (int f = 0; f < FDIM; ++f)
        man[(size_t)row * FDIM + f] = gelu_exact((v[f] - mu) * rs * g[f] + b[f]);
}

// ---------------------------------------------------------------------------
// K3: one Euler ODE step: man += dt * (tanh(man@w1+b1) @ w2 + b2)
// ---------------------------------------------------------------------------
__global__ void __launch_bounds__(128) ode_step(float* __restrict__ man,
                                                const float* __restrict__ w1,
                                                const float* __restrict__ b1,
                                                const float* __restrict__ w2,
                                                const float* __restrict__ b2) {
    __shared__ float srow[FDIM];
    __shared__ float sh[HODE];
    const int tid = threadIdx.x;
    const int tok = blockIdx.x;
    if (tid < FDIM) srow[tid] = man[(size_t)tok * FDIM + tid];
    __syncthreads();
    float hj = b1[tid];
    for (int k = 0; k < FDIM; ++k) hj += srow[k] * w1[k * HODE + tid];
    sh[tid] = tanhf(hj);
    __syncthreads();
    if (tid < FDIM) {
        float d = b2[tid];
        for (int j = 0; j < HODE; ++j) d += sh[j] * w2[j * FDIM + tid];
        man[(size_t)tok * FDIM + tid] = srow[tid] + DTSTEP * d;
    }
}

// ---------------------------------------------------------------------------
// K3b: per-slot ||mem_pos||^2 and curvature weight exp(-alpha*||curv||)
// ---------------------------------------------------------------------------
__global__ void __launch_bounds__(256) slot_stats(const float* __restrict__ pos,
                                                  const float* __restrict__ curv,
                                                  const float* __restrict__ alpha_p,
                                                  float* __restrict__ mm,
                                                  float* __restrict__ cw) {
    const int m = blockIdx.x * blockDim.x + threadIdx.x;
    float s = 0.f;
    for (int f = 0; f < FDIM; ++f) { float p = pos[(size_t)m * FDIM + f]; s += p * p; }
    mm[m] = s;
    float c = 0.f;
    for (int d = 0; d < 16; ++d) { float q = curv[m * 16 + d]; c += q * q; }
    cw[m] = expf(-alpha_p[0] * sqrtf(c));
}

// ---------------------------------------------------------------------------
// K4: fused distance GEMM (WMMA f32) + per-token top-48 (LDS max-heap)
//     block = 8 waves = 16 tokens; slots streamed in 128-wide chunks.
//     sq rows padded to 50 floats -> conflict-free A-fragment LDS reads.
//     Next chunk of mem_pos prefetched (global_prefetch_b8) during select.
// ---------------------------------------------------------------------------
#define FPAD (FDIM + 2)
__global__ void __launch_bounds__(256) dist_topk(const float* __restrict__ man,
                                                 const float* __restrict__ pos,
                                                 const float* __restrict__ mm,
                                                 const float* __restrict__ cw,
                                                 float* __restrict__ dtop,
                                                 int*   __restrict__ itop,
                                                 float* __restrict__ top1) {
    __shared__ float sq[16][FPAD];
    __shared__ float sqq[16];
    __shared__ float sdist[16][128];
    __shared__ float hd[16][KBIG];
    __shared__ int   hi[16][KBIG];
    const int tid  = threadIdx.x;
    const int lane = tid & 31, wave = tid >> 5;
    const int half = lane >> 4, lm = lane & 15;
    const int token0 = blockIdx.x * 16;

    for (int i = tid; i < 16 * FDIM; i += 256)
        sq[i / FDIM][i % FDIM] = man[(size_t)(token0 + i / FDIM) * FDIM + (i % FDIM)];
    for (int i = tid; i < 16 * KBIG; i += 256) {
        hd[i / KBIG][i % KBIG] = 3.0e30f;
        hi[i / KBIG][i % KBIG] = 0;
    }
    __syncthreads();
    if (tid < 16) {
        float s = 0.f;
        for (int f = 0; f < FDIM; ++f) s += sq[tid][f] * sq[tid][f];
        sqq[tid] = s;
    }
    __syncthreads();

    for (int chunk = 0; chunk < MSLOTS; chunk += 128) {
        const int slot0 = chunk + wave * 16;
        v8f acc = {0.f,0.f,0.f,0.f,0.f,0.f,0.f,0.f};
        for (int k = 0; k < FDIM; k += 4) {
            const int ka = k + 2 * half;
            v2f a = *(const v2f*)&sq[lm][ka];
            v2f b = *(const v2f*)(pos + (size_t)(slot0 + lm) * FDIM + ka);
            acc = wmma_f32(a, b, acc);
        }
        {
            const int sl = wave * 16 + lm;
            const int sg = chunk + sl;
            const float mmv = mm[sg];
            const float cwv = cw[sg];
#pragma unroll
            for (int r = 0; r < 8; ++r) {
                const int tok = r + 8 * half;
                float d2 = sqq[tok] + mmv - 2.0f * acc[r];
                sdist[tok][sl] = sqrtf(fmaxf(d2, 0.f)) * cwv;
            }
        }
        // prefetch next chunk's slot rows into cache while selection runs
        if (chunk + 128 < MSLOTS) {
            const int ps = chunk + 128 + (tid >> 1);           // 128 rows / 256 thr
            __builtin_prefetch(pos + (size_t)ps * FDIM + (tid & 1) * 24, 0, 1);
        }
        __syncthreads();
        if (tid < 16) {
            float* hdt = hd[tid];
            int*   hit = hi[tid];
            for (int j = 0; j < 128; ++j) {
                const float d = sdist[tid][j];
                if (d < hdt[0]) {                 // replace max root, sift down
                    hdt[0] = d; hit[0] = chunk + j;
                    int i = 0;
                    while (true) {
                        int l = 2 * i + 1, r2 = 2 * i + 2, big = i;
                        if (l  < KBIG && hdt[l]  > hdt[big]) big = l;
                        if (r2 < KBIG && hdt[r2] > hdt[big]) big = r2;
                        if (big == i) break;
                        float td = hdt[i]; hdt[i] = hdt[big]; hdt[big] = td;
                        int   ti = hit[i]; hit[i] = hit[big]; hit[big] = ti;
                        i = big;
                    }
                }
            }
        }
        __syncthreads();
    }

    if (tid < 16) {   // ascending sort -> rank order matches jax top_k
        float* hdt = hd[tid];
        int*   hit = hi[tid];
        for (int i = 0; i < KBIG; ++i) {
            int mn = i;
            for (int j = i + 1; j < KBIG; ++j) if (hdt[j] < hdt[mn]) mn = j;
            float td = hdt[i]; hdt[i] = hdt[mn]; hdt[mn] = td;
            int   ti = hit[i]; hit[i] = hit[mn]; hit[mn] = ti;
            dtop[(size_t)(token0 + tid) * KBIG + i] = hdt[i];
            itop[(size_t)(token0 + tid) * KBIG + i] = hit[i];
        }
        top1[token0 + tid] = hdt[0];
    }
}

// ---------------------------------------------------------------------------
// K5: deterministic reduction of per-token top1 -> fire flag
// ---------------------------------------------------------------------------
__global__ void __launch_bounds__(256) fire_reduce(const float* __restrict__ top1,
                                                   int* __restrict__ fire) {
    __shared__ float red[256];
    const int tid = threadIdx.x;
    float s = 0.f;
    for (int i = tid; i < TOKENS; i += 256) s += top1[i];
    red[tid] = s;
    __syncthreads();
    for (int off = 128; off > 0; off >>= 1) {
        if (tid < off) red[tid] += red[tid + off];
        __syncthreads();
    }
    if (tid == 0) fire[0] = (red[0] * (1.0f / TOKENS) < LB_TH) ? 1 : 0;
}

// ---------------------------------------------------------------------------
// K6: masked softmax over -dtop + gather-weighted sum of memory slots
// ---------------------------------------------------------------------------
__global__ void __launch_bounds__(256) attend(const float* __restrict__ dtop,
                                              const int*   __restrict__ itop,
                                              const int*   __restrict__ fire,
                                              const float* __restrict__ mem,
                                              float* __restrict__ att) {
    __shared__ float wk[KBIG];
    __shared__ int   sidx[KBIG];
    __shared__ float ssum;
    const int tid = threadIdx.x;
    const int tok = blockIdx.x;
    if (tid < KBIG) {
        const float d  = dtop[(size_t)tok * KBIG + tid];
        const float d0 = dtop[(size_t)tok * KBIG];     // min dist; softmax max = -d0
        const int keep = fire[0] | ((tid < KBASE) ? 1 : 0);
        wk[tid]   = keep ? expf(d0 - d) : 0.f;
        sidx[tid] = itop[(size_t)tok * KBIG + tid];
    }
    __syncthreads();
    if (tid == 0) {
        float s = 0.f;
        for (int k = 0; k < KBIG; ++k) s += wk[k];
        ssum = s;
    }
    __syncthreads();
    const float inv = 1.0f / ssum;
    float acc = 0.f;
    for (int k = 0; k < KBIG; ++k)
        acc += wk[k] * mem[(size_t)sidx[k] * HDIM + tid];
    att[(size_t)tok * HDIM + tid] = acc * inv;
}

// ---------------------------------------------------------------------------
// K7: out = attended @ w_out + b_out  (fp32 WMMA), writes straight to d_out.
//     Block = 8 waves owns one 16-row M-tile; the 16x256 A panel is staged
//     once in LDS (padded rows -> conflict-free fragment reads); each wave
//     sweeps 8 N-tiles in pairs sharing one A-fragment read per 2 WMMAs.
// ---------------------------------------------------------------------------
#define HPAD (HDIM + 2)
__global__ void __launch_bounds__(256) out_gemm(const float* __restrict__ att,
                                                const float* __restrict__ wo,
                                                const float* __restrict__ bo,
                                                float* __restrict__ out) {
    __shared__ float sa[16][HPAD];
    const int tid  = threadIdx.x;
    const int lane = tid & 31, wave = tid >> 5;
    const int half = lane >> 4, lm = lane & 15;
    const int m0 = blockIdx.x * 16;
    for (int i = tid; i < 16 * HDIM; i += 256)
        sa[i / HDIM][i % HDIM] = att[(size_t)(m0 + i / HDIM) * HDIM + (i % HDIM)];
    __syncthreads();
#pragma unroll
    for (int p = 0; p < 4; ++p) {                 // 4 pairs of N-tiles per wave
        const int n0 = (wave * 8 + p * 2) * 16;
        v8f acc0 = {0.f,0.f,0.f,0.f,0.f,0.f,0.f,0.f};
        v8f acc1 = acc0;
        for (int k = 0; k < HDIM; k += 4) {
            const int ka = k + 2 * half;
            v2f a = *(const v2f*)&sa[lm][ka];
            const float* wr0 = wo + (size_t)ka * IN_DIM + n0 + lm;
            const float* wr1 = wr0 + IN_DIM;
            v2f b0, b1;
            b0.x = wr0[0];  b0.y = wr1[0];
            b1.x = wr0[16]; b1.y = wr1[16];
            acc0 = wmma_f32(a, b0, acc0);
            acc1 = wmma_f32(a, b1, acc1);
        }
#pragma unroll
        for (int r = 0; r < 8; ++r) {
            const int row = m0 + r + 8 * half;
            float* orow = out + (size_t)row * IN_DIM + n0;
            orow[lm]      = acc0[r] + bo[n0 + lm];
            orow[lm + 16] = acc1[r] + bo[n0 + lm + 16];
        }
    }
}

// ---------------------------------------------------------------------------
// K8: LayerNorm(1024) + exact GELU, in place on d_out (one block per row)
// ---------------------------------------------------------------------------
__global__ void __launch_bounds__(256) ln_gelu_out(float* __restrict__ out,
                                                   const float* __restrict__ g,
                                                   const float* __restrict__ b) {
    __shared__ float red[256];
    const int tid = threadIdx.x;
    float* row = out + (size_t)blockIdx.x * IN_DIM;
    float v[4];
    float s = 0.f;
#pragma unroll
    for (int i = 0; i < 4; ++i) { v[i] = row[tid + 256 * i]; s += v[i]; }
    red[tid] = s;
    __syncthreads();
    for (int off = 128; off > 0; off >>= 1) {
        if (tid < off) red[tid] += red[tid + off];
        __syncthreads();
    }
    const float mu = red[0] * (1.0f / IN_DIM);
    __syncthreads();                    // WAR guard before reusing red[]
    s = 0.f;
#pragma unroll
    for (int i = 0; i < 4; ++i) { float d = v[i] - mu; s += d * d; }
    red[tid] = s;
    __syncthreads();
    for (int off = 128; off > 0; off >>= 1) {
        if (tid < off) red[tid] += red[tid + off];
        __syncthreads();
    }
    const float rs = rsqrtf(red[0] * (1.0f / IN_DIM) + 1e-5f);
#pragma unroll
    for (int i = 0; i < 4; ++i) {
        const int c = tid + 256 * i;
        row[c] = gelu_exact((v[i] - mu) * rs * g[c] + b[c]);
    }
}

// ---------------------------------------------------------------------------
extern "C" void kernel_launch(void* const* d_in, const int* in_sizes, int n_in,
                              void* d_out, int out_size, void* d_ws, size_t ws_size,
                              hipStream_t stream) {
    (void)in_sizes; (void)n_in; (void)out_size; (void)ws_size;
    const float* x     = (const float*)d_in[0];
    const float* wproj = (const float*)d_in[1];
    const float* bproj = (const float*)d_in[2];
    const float* ln1g  = (const float*)d_in[3];
    const float* ln1b  = (const float*)d_in[4];
    const float* w1    = (const float*)d_in[5];
    const float* b1    = (const float*)d_in[6];
    const float* w2    = (const float*)d_in[7];
    const float* b2    = (const float*)d_in[8];
    const float* mem   = (const float*)d_in[9];
    const float* pos   = (const float*)d_in[10];   // (M,16,3) == (M,48) flat
    const float* curv  = (const float*)d_in[11];
    const float* alpha = (const float*)d_in[12];
    const float* wout  = (const float*)d_in[13];
    const float* bout  = (const float*)d_in[14];
    const float* ln2g  = (const float*)d_in[15];
    const float* ln2b  = (const float*)d_in[16];
    float* out = (float*)d_out;

    char* ws = (char*)d_ws;
    float* man  = (float*)(ws + 0);          // 8192*48*4  = 1,572,864
    float* mm   = (float*)(ws + 1572864);    // 32 KB
    float* cw   = (float*)(ws + 1605632);    // 32 KB
    float* dtop = (float*)(ws + 1638400);    // 1,572,864
    int*   itop = (int*)  (ws + 3211264);    // 1,572,864
    float* top1 = (float*)(ws + 4784128);    // 32 KB
    int*   fire = (int*)  (ws + 4816896);    // 256 B
    float* att  = (float*)(ws + 4817152);    // 8 MB  (total ~13.2 MB)

    proj_gemm <<<TOKENS / 64, 128, 0, stream>>>(x, wproj, bproj, man);
    ln_gelu_f <<<TOKENS / 256, 256, 0, stream>>>(man, ln1g, ln1b);
    slot_stats<<<MSLOTS / 256, 256, 0, stream>>>(pos, curv, alpha, mm, cw);
    ode_step  <<<TOKENS, 128, 0, stream>>>(man, w1, b1, w2, b2);
    ode_step  <<<TOKENS, 128, 0, stream>>>(man, w1, b1, w2, b2);
    dist_topk <<<TOKENS / 16, 256, 0, stream>>>(man, pos, mm, cw, dtop, itop, top1);
    fire_reduce<<<1, 256, 0, stream>>>(top1, fire);
    attend    <<<TOKENS, 256, 0, stream>>>(dtop, itop, fire, mem, att);
    out_gemm  <<<TOKENS / 16, 256, 0, stream>>>(att, wout, bout, out);
    ln_gelu_out<<<TOKENS, 256, 0, stream>>>(out, ln2g, ln2b);
}